// HGNN_gs_block_17497696764598
// MI455X (gfx1250) — compile-verified
//
#include <hip/hip_runtime.h>

typedef float v2f __attribute__((ext_vector_type(2)));
typedef float v8f __attribute__((ext_vector_type(8)));

#define N_NODES 4096
#define F_DIM   256
#define E_DIM   128
#define K_NEIG  9
#define NT      4      // 16x16 N-tiles per wave -> 16x64 output per wave

// ---------------------------------------------------------------------------
// 1) Top-k per row (k=9 largest, diagonal forced to 0, center forced into the
//    last slot if absent). One wave (32 lanes) per row, 9 argmax passes with
//    tie -> smallest index (matches lax.top_k stability).
//    Also accumulates node degrees DV via float atomics.
// ---------------------------------------------------------------------------
__global__ void topk_rows_kernel(const float* __restrict__ dis,
                                 int* __restrict__ idxOut,
                                 float* __restrict__ DV) {
    const int row  = blockIdx.x;
    const int lane = threadIdx.x;            // 0..31 (wave32)
    const float* rowp = dis + (size_t)row * N_NODES;

    int sel[K_NEIG];
    for (int p = 0; p < K_NEIG; ++p) {
        float best = -1.0f;                   // all values in [0,1)
        int   bidx = N_NODES;
        for (int c = lane; c < N_NODES; c += 32) {
            float v = (c == row) ? 0.0f : rowp[c];
            bool skip = false;
            for (int q = 0; q < p; ++q) skip |= (sel[q] == c);
            if (!skip && (v > best || (v == best && c < bidx))) {
                best = v; bidx = c;
            }
        }
        // wave32 argmax reduction, tie -> smaller index
        for (int off = 16; off > 0; off >>= 1) {
            float ov = __shfl_xor(best, off, 32);
            int   oi = __shfl_xor(bidx, off, 32);
            if (ov > best || (ov == best && oi < bidx)) { best = ov; bidx = oi; }
        }
        sel[p] = bidx;                        // uniform across the wave
    }

    bool has_center = false;
    for (int q = 0; q < K_NEIG; ++q) has_center |= (sel[q] == row);
    if (!has_center) sel[K_NEIG - 1] = row;

    if (lane == 0) {
        for (int q = 0; q < K_NEIG; ++q) idxOut[row * K_NEIG + q] = sel[q];
    }
    if (lane < K_NEIG) atomicAdd(&DV[sel[lane]], 1.0f);
}

// ---------------------------------------------------------------------------
// 2) dv2[i] = DV[i]^-1/2
// ---------------------------------------------------------------------------
__global__ void dv2_kernel(const float* __restrict__ DV, float* __restrict__ dv2) {
    int i = blockIdx.x * blockDim.x + threadIdx.x;
    if (i < N_NODES) dv2[i] = 1.0f / sqrtf(DV[i]);
}

// ---------------------------------------------------------------------------
// 3) Sparse scatter of G: for hyperedge e with members idx[e][0..8],
//    G[i,j] += dv2[i]*dv2[j] / DE  (DE == 9 exactly).
//    G has <= ~81 nnz/row analytically, so scatter (330K atomics) beats a
//    dense 4096^3 H@H^T GEMM by ~400x in work.
// ---------------------------------------------------------------------------
__global__ void g_scatter_kernel(const int* __restrict__ idx,
                                 const float* __restrict__ dv2,
                                 float* __restrict__ G) {
    const int e = blockIdx.x;
    const int t = threadIdx.x;
    if (t >= K_NEIG * K_NEIG) return;
    const int a = t / K_NEIG, b = t % K_NEIG;
    const int i = idx[e * K_NEIG + a];
    const int j = idx[e * K_NEIG + b];
    const float w = dv2[i] * dv2[j] * (1.0f / (float)K_NEIG);
    atomicAdd(&G[(size_t)i * N_NODES + j], w);
}

// ---------------------------------------------------------------------------
// 4) fp32 WMMA GEMM: C[M,Nc] = opt_relu(A[M,Kd] @ B[Kd,Nc] + bias).
//    One wave computes a 16(M) x 64(N) strip: 4 accumulators share one
//    A-fragment per k-step (4x less A traffic, 4 wmma per A load).
//    V_WMMA_F32_16X16X4_F32 fragment layouts per CDNA5 ISA 7.12.2:
//      A 16x4 : lane<16 -> (M=lane, K=k0+{0,1}); lane>=16 -> (M=lane-16, K=k0+{2,3})
//      B 4x16 : lane<16 -> (K=k0+{0,1}, N=lane); lane>=16 -> (K=k0+{2,3}, N=lane-16)
//      C 16x16: VGPR r, lane l -> (M = r + 8*(l>=16), N = l%16)
// ---------------------------------------------------------------------------
__global__ void wmma_gemm_f32_kernel(const float* __restrict__ A,
                                     const float* __restrict__ B,
                                     const float* __restrict__ bias,
                                     float* __restrict__ C,
                                     int M, int Nc, int Kd, int do_relu) {
    const int tm   = blockIdx.x;             // M/16 tiles
    const int tn   = blockIdx.y;             // Nc/64 strips
    const int lane = threadIdx.x;            // 0..31
    const int half = lane >> 4;              // 0 or 1
    const int l16  = lane & 15;

    const int rowA  = tm * 16 + l16;
    const int colB0 = tn * (16 * NT) + l16;  // this lane's column in N-tile 0
    const float* __restrict__ ap = A + (size_t)rowA * Kd;

    v8f acc[NT];
    #pragma unroll
    for (int j = 0; j < NT; ++j) acc[j] = (v8f){0.f,0.f,0.f,0.f,0.f,0.f,0.f,0.f};

    #pragma unroll 2
    for (int k0 = 0; k0 < Kd; k0 += 4) {
        const int ka = k0 + half * 2;
        v2f a;
        a.x = ap[ka];
        a.y = ap[ka + 1];
        const float* __restrict__ bp0 = B + (size_t)ka * Nc + colB0;
        const float* __restrict__ bp1 = bp0 + Nc;
        #pragma unroll
        for (int j = 0; j < NT; ++j) {
            v2f b;
            b.x = bp0[j * 16];
            b.y = bp1[j * 16];
            acc[j] = __builtin_amdgcn_wmma_f32_16x16x4_f32(
                /*neg_a=*/false, a, /*neg_b=*/false, b,
                /*c_mod=*/(short)0, acc[j], /*reuse_a=*/false, /*reuse_b=*/false);
        }
    }

    const int baseM = tm * 16 + half * 8;
    #pragma unroll
    for (int j = 0; j < NT; ++j) {
        const int col = colB0 + j * 16;
        const float bval = bias ? bias[col] : 0.0f;
        #pragma unroll
        for (int r = 0; r < 8; ++r) {
            float v = acc[j][r] + bval;
            if (do_relu) v = fmaxf(v, 0.0f);
            C[(size_t)(baseM + r) * Nc + col] = v;
        }
    }
}

// ---------------------------------------------------------------------------
// Host-side orchestration
// ---------------------------------------------------------------------------
extern "C" void kernel_launch(void* const* d_in, const int* in_sizes, int n_in,
                              void* d_out, int out_size, void* d_ws, size_t ws_size,
                              hipStream_t stream) {
    const float* x   = (const float*)d_in[0];   // [4096,256]
    const float* XY  = (const float*)d_in[1];   // [4096,4096]
    const float* W10 = (const float*)d_in[2];   // [256,128]
    const float* b10 = (const float*)d_in[3];   // [128]
    const float* W11 = (const float*)d_in[4];   // [128,128]
    const float* b11 = (const float*)d_in[5];   // [128]
    const float* W20 = (const float*)d_in[6];   // [128,128]
    const float* b20 = (const float*)d_in[7];   // [128]
    const float* W21 = (const float*)d_in[8];   // [128,256]
    const float* b21 = (const float*)d_in[9];   // [256]

    float* out_x  = (float*)d_out;                        // [4096,256]
    float* out_x1 = out_x  + (size_t)N_NODES * F_DIM;     // [4096,128]
    float* out_x2 = out_x1 + (size_t)N_NODES * E_DIM;     // [4096,256]

    // Workspace layout (fp32 unless noted)
    float* G   = (float*)d_ws;                               // 4096*4096
    float* DV  = G  + (size_t)N_NODES * N_NODES;             // 4096
    float* dv2 = DV + N_NODES;                               // 4096
    int*   idx = (int*)(dv2 + N_NODES);                      // 4096*9 ints
    float* t0  = (float*)(idx + (size_t)N_NODES * 12);       // 4096*256 (padded idx)
    float* t1  = t0 + (size_t)N_NODES * F_DIM;               // 4096*256

    // Zero G and DV
    hipMemsetAsync(G,  0, (size_t)N_NODES * N_NODES * sizeof(float), stream);
    hipMemsetAsync(DV, 0, (size_t)N_NODES * sizeof(float), stream);

    // Build hyperedges + degrees
    topk_rows_kernel<<<N_NODES, 32, 0, stream>>>(XY, idx, DV);
    dv2_kernel<<<(N_NODES + 255) / 256, 256, 0, stream>>>(DV, dv2);
    g_scatter_kernel<<<N_NODES, 96, 0, stream>>>(idx, dv2, G);

    // out[0] = x (identity passthrough)
    hipMemcpyAsync(out_x, x, (size_t)N_NODES * F_DIM * sizeof(float),
                   hipMemcpyDeviceToDevice, stream);

    dim3 wave(32, 1, 1);
    dim3 g128(N_NODES / 16, E_DIM / (16 * NT), 1);   // [4096,128] outputs
    dim3 g256(N_NODES / 16, F_DIM / (16 * NT), 1);   // [4096,256] outputs

    // ---- layer 1 ----
    // t0 = x @ W10 + b10
    wmma_gemm_f32_kernel<<<g128, wave, 0, stream>>>(x,  W10, b10, t0, N_NODES, E_DIM, F_DIM, 0);
    // t1 = relu(G @ t0)
    wmma_gemm_f32_kernel<<<g128, wave, 0, stream>>>(G,  t0, nullptr, t1, N_NODES, E_DIM, N_NODES, 1);
    // t0 = t1 @ W11 + b11
    wmma_gemm_f32_kernel<<<g128, wave, 0, stream>>>(t1, W11, b11, t0, N_NODES, E_DIM, E_DIM, 0);
    // x1 = relu(G @ t0)
    wmma_gemm_f32_kernel<<<g128, wave, 0, stream>>>(G,  t0, nullptr, out_x1, N_NODES, E_DIM, N_NODES, 1);

    // ---- layer 2 ----
    // t0 = x1 @ W20 + b20
    wmma_gemm_f32_kernel<<<g128, wave, 0, stream>>>(out_x1, W20, b20, t0, N_NODES, E_DIM, E_DIM, 0);
    // t1 = relu(G @ t0)
    wmma_gemm_f32_kernel<<<g128, wave, 0, stream>>>(G,  t0, nullptr, t1, N_NODES, E_DIM, N_NODES, 1);
    // t0 = t1 @ W21 + b21  -> [4096,256]
    wmma_gemm_f32_kernel<<<g256, wave, 0, stream>>>(t1, W21, b21, t0, N_NODES, F_DIM, E_DIM, 0);
    // x2 = relu(G @ t0)
    wmma_gemm_f32_kernel<<<g256, wave, 0, stream>>>(G,  t0, nullptr, out_x2, N_NODES, F_DIM, N_NODES, 1);
}